// dgl_appnp_73529840107895
// MI455X (gfx1250) — compile-verified
//
#include <hip/hip_runtime.h>
#include <math.h>

#define ALPHA 0.1f

typedef float v2f __attribute__((ext_vector_type(2)));
typedef float v8f __attribute__((ext_vector_type(8)));

// ---------------- hardware fp32 global atomic add (non-returning path) ----------------
__device__ __forceinline__ void atomic_add_f32(float* p, float v) {
#if __has_builtin(__builtin_amdgcn_global_atomic_fadd_f32)
  typedef __attribute__((address_space(1))) float gfloat;
  __builtin_amdgcn_global_atomic_fadd_f32((gfloat*)p, v);
#else
  atomicAdd(p, v);
#endif
}

// ---------------- elementwise helpers ----------------

__global__ void zero4_k(float4* __restrict__ p, int n4) {
  int i = blockIdx.x * blockDim.x + threadIdx.x;
  if (i < n4) p[i] = make_float4(0.f, 0.f, 0.f, 0.f);
}

__global__ void deg_k(const int* __restrict__ dst, float* __restrict__ deg, int E) {
  int e = blockIdx.x * blockDim.x + threadIdx.x;
  if (e < E) atomic_add_f32(&deg[dst[e]], 1.0f);
}

__global__ void norm_k(float* __restrict__ nrm, int n) {
  int i = blockIdx.x * blockDim.x + threadIdx.x;
  if (i < n) {
    float d = nrm[i];
    nrm[i] = 1.0f / sqrtf(fmaxf(d, 1.0f));
  }
}

__device__ __forceinline__ float elu1(float v) {
  return v > 0.0f ? v : (expf(v) - 1.0f);
}

// ---------------- WMMA GEMM: C[N x HID] = A[N x KDIM] * W[HID x KDIM]^T + bias ----------------
// One wave computes one 16x16 output tile with V_WMMA_F32_16X16X4_F32.
// A-frag (16x4 fp32, ISA 7.12.2): lanes 0-15 hold M rows, vgpr0/1 = K=k,k+1;
//                                 lanes 16-31 same rows, vgpr0/1 = K=k+2,k+3.
// B-frag (4x16 fp32): lanes = N cols, vgpr0 = row K=k (lanes 0-15) / K=k+2 (16-31),
//                                     vgpr1 = row K=k+1 / K=k+3.  B[k][n] = W[n][k].
// C/D: vgpr v -> (m = v + 8*(lane>=16), n = lane&15).
template <int KDIM, int HID, bool ELU>
__global__ void gemm_wmma(const float* __restrict__ A, const float* __restrict__ W,
                          const float* __restrict__ bias, float* __restrict__ C, int N) {
  constexpr int NTN = HID / 16;
  const int tiles = ((N + 15) / 16) * NTN;
  // Force the per-wave tile index into an SGPR so tm/tn and all guards are scalar
  // (s_cbranch instead of exec-masked dual paths); it is wave-uniform by construction.
  const int tile = __builtin_amdgcn_readfirstlane(
      blockIdx.x * (blockDim.x >> 5) + (threadIdx.x >> 5));
  if (tile >= tiles) return;  // scalar branch; EXEC stays all-1s for WMMA below
  const int tm = tile / NTN;
  const int tn = tile % NTN;
  const int lane = threadIdx.x & 31;
  const int half = lane >> 4;     // 0 = lanes 0-15, 1 = lanes 16-31
  const int l15 = lane & 15;
  const int row = tm * 16 + l15;
  const int rowc = row < N ? row : N - 1;   // clamp tail loads
  const int col = tn * 16 + l15;

  v8f acc = {};
  for (int k = 0; k < KDIM; k += 4) {
    const int ka = k + 2 * half;
    v2f a, b;
    a.x = A[rowc * KDIM + ka];
    a.y = A[rowc * KDIM + ka + 1];
    b.x = W[col * KDIM + ka];
    b.y = W[col * KDIM + ka + 1];
    acc = __builtin_amdgcn_wmma_f32_16x16x4_f32(false, a, false, b,
                                                (short)0, acc, false, false);
  }
  const float bs = bias[col];
  if (tm * 16 + 15 < N) {  // scalar condition (tm is SGPR): full tile, unguarded stores
#pragma unroll
    for (int v = 0; v < 8; ++v) {
      int m = tm * 16 + v + 8 * half;
      float x = acc[v] + bs;
      if (ELU) x = elu1(x);
      C[m * HID + col] = x;
    }
  } else {
#pragma unroll
    for (int v = 0; v < 8; ++v) {
      int m = tm * 16 + v + 8 * half;
      if (m < N) {
        float x = acc[v] + bs;
        if (ELU) x = elu1(x);
        C[m * HID + col] = x;
      }
    }
  }
}

// ---------------- APPNP propagation ----------------
// 16 lanes per edge; lane group covers D channels (D multiple of 16).
// Gathers hit L2 (h fits: <=19.2 MB); scatter uses hardware global_atomic_add_f32.
template <int D>
__global__ void edge_k(const int* __restrict__ src, const int* __restrict__ dst,
                       const float* __restrict__ h, const float* __restrict__ nrm,
                       float* __restrict__ agg, int E) {
  int t = blockIdx.x * blockDim.x + threadIdx.x;
  int e = t >> 4;
  int c = t & 15;
  if (e >= E) return;
  int s = src[e];
  int d = dst[e];
  float w = nrm[s];
#pragma unroll
  for (int j = 0; j < D / 16; ++j) {
    int ch = c + 16 * j;
    atomic_add_f32(&agg[d * D + ch], h[s * D + ch] * w);
  }
}

// h_new = (1-alpha) * agg * norm[node] + alpha * h0, float4-wide.
// Valid because 4 divides D, so a float4 never crosses a node-row boundary.
// ELU=true additionally applies elu and writes to `out` (used for the final step).
template <int D, bool ELU>
__global__ void combine4_k(const float* __restrict__ nrm, const float4* __restrict__ h0,
                           float4* __restrict__ agg, float4* __restrict__ out, int N) {
  int i = blockIdx.x * blockDim.x + threadIdx.x;
  if (i >= N * (D / 4)) return;
  int node = i / (D / 4);
  float w = (1.0f - ALPHA) * nrm[node];
  float4 a = agg[i];
  float4 h = h0[i];
  float4 r = make_float4(w * a.x + ALPHA * h.x, w * a.y + ALPHA * h.y,
                         w * a.z + ALPHA * h.z, w * a.w + ALPHA * h.w);
  if (ELU) {
    out[i] = make_float4(elu1(r.x), elu1(r.y), elu1(r.z), elu1(r.w));
  } else {
    agg[i] = r;
  }
}

// ---------------- driver ----------------

extern "C" void kernel_launch(void* const* d_in, const int* in_sizes, int n_in,
                              void* d_out, int out_size, void* d_ws, size_t ws_size,
                              hipStream_t stream) {
  const float* F   = (const float*)d_in[0];
  const int*   src = (const int*)d_in[1];
  const int*   dst = (const int*)d_in[2];
  const float* W1  = (const float*)d_in[3];
  const float* b1  = (const float*)d_in[4];
  const float* W2  = (const float*)d_in[5];
  const float* b2  = (const float*)d_in[6];
  float* out = (float*)d_out;

  constexpr int IN = 256, HID = 48, CLS = 16, KSTEPS = 10;
  const int N = in_sizes[0] / IN;
  const int E = in_sizes[1];

  // Workspace layout (floats): norm[N] | h0[N*48] | hA[N*48] | hB[N*48]
  // D=16 phase reuses h0's region (dead after propagation 1): x0|xA|xB fit in N*48.
  float* ws  = (float*)d_ws;
  float* nrm = ws;
  float* h0  = nrm + N;
  float* hA  = h0 + (size_t)N * HID;
  float* hB  = hA + (size_t)N * HID;
  float* x0  = h0;
  float* xA  = x0 + (size_t)N * CLS;
  float* xB  = xA + (size_t)N * CLS;

  auto blocks = [](long n, int t) { return (int)((n + t - 1) / t); };

  // degree -> norm (N divisible by 4 here; zero via float4)
  zero4_k<<<blocks(N / 4, 256), 256, 0, stream>>>((float4*)nrm, N / 4);
  deg_k<<<blocks(E, 256), 256, 0, stream>>>(dst, nrm, E);
  norm_k<<<blocks(N, 256), 256, 0, stream>>>(nrm, N);

  // fc1 (WMMA fp32)
  const int tiles1 = ((N + 15) / 16) * (HID / 16);
  gemm_wmma<IN, HID, false><<<blocks((long)tiles1 * 32, 128), 128, 0, stream>>>(F, W1, b1, h0, N);

  // propagation 1, D=48
  const float* cur = h0;
  for (int t = 0; t < KSTEPS; ++t) {
    float* agg = (t & 1) ? hB : hA;
    zero4_k<<<blocks((long)N * HID / 4, 256), 256, 0, stream>>>((float4*)agg, N * (HID / 4));
    edge_k<HID><<<blocks((long)E * 16, 256), 256, 0, stream>>>(src, dst, cur, nrm, agg, E);
    combine4_k<HID, false><<<blocks((long)N * HID / 4, 256), 256, 0, stream>>>(
        nrm, (const float4*)h0, (float4*)agg, nullptr, N);
    cur = agg;
  }

  // fc2 + ELU (WMMA fp32); cur == hB after 10 steps
  const int tiles2 = ((N + 15) / 16) * (CLS / 16);
  gemm_wmma<HID, CLS, true><<<blocks((long)tiles2 * 32, 128), 128, 0, stream>>>(cur, W2, b2, x0, N);

  // propagation 2, D=16; final step fuses combine + ELU and writes d_out
  const float* cur2 = x0;
  for (int t = 0; t < KSTEPS; ++t) {
    float* agg = (t & 1) ? xB : xA;
    zero4_k<<<blocks((long)N * CLS / 4, 256), 256, 0, stream>>>((float4*)agg, N * (CLS / 4));
    edge_k<CLS><<<blocks((long)E * 16, 256), 256, 0, stream>>>(src, dst, cur2, nrm, agg, E);
    if (t == KSTEPS - 1) {
      combine4_k<CLS, true><<<blocks((long)N * CLS / 4, 256), 256, 0, stream>>>(
          nrm, (const float4*)x0, (float4*)agg, (float4*)out, N);
    } else {
      combine4_k<CLS, false><<<blocks((long)N * CLS / 4, 256), 256, 0, stream>>>(
          nrm, (const float4*)x0, (float4*)agg, nullptr, N);
    }
    cur2 = agg;
  }
}